// PyG_PointTransformerSegModel_8727373545753
// MI455X (gfx1250) — compile-verified
//
#include <hip/hip_runtime.h>
#include <hip/hip_bf16.h>

typedef __attribute__((ext_vector_type(2))) float v2f;
typedef __attribute__((ext_vector_type(8))) float v8f;

// ---------------------------------------------------------------------------
// fp32 WMMA GEMM:  C[M,N] = A[M,K] @ B[K,N] (+ bias), V_WMMA_F32_16X16X4_F32.
//
// Block = 256 threads = 8 waves. Each block covers a 128-row strip of A and
// the FULL (zero-padded) N width. The whole B panel (K x NP floats, <=64KB)
// is cooperatively staged into LDS once, swizzled so that the two K-adjacent
// elements of a B fragment are contiguous:
//     B(k, n)  ->  lds[(k>>1)*(2*NP) + n*2 + (k&1)]
// so each fragment load is a single 8-byte ds_load_b64 and the K-loop is
// completely branch-free. N-padding handled at stage (zeros) + store (guard).
//
// Fragment layouts (CDNA5 ISA):
//  A (16x4):  lane<16: A[m=lane][k0+0..1]; lane>=16: A[m=lane-16][k0+2..3]
//  B (4x16):  lane<16: B[k0+0..1][n=lane]; lane>=16: B[k0+2..3][n=lane-16]
//  C (16x16): VGPR i -> row m0+i (lanes 0-15) / m0+8+i (lanes 16-31),
//             col = n0 + (lane&15)
// ---------------------------------------------------------------------------
template <int NTILES>
__global__ void wmma_gemm_f32(const float* __restrict__ A,
                              const float* __restrict__ B,
                              const float* __restrict__ bias,
                              float* __restrict__ C,
                              int M, int N, int K) {
  extern __shared__ float ldsB[];
  const int NP = NTILES * 16;

  // ---- stage zero-padded, swizzled B panel into LDS
  for (int idx = threadIdx.x; idx < K * NP; idx += blockDim.x) {
    int k = idx / NP;
    int n = idx - k * NP;
    float val = (n < N) ? B[(size_t)k * N + n] : 0.0f;
    ldsB[(k >> 1) * (2 * NP) + n * 2 + (k & 1)] = val;
  }
  __syncthreads();

  const int lane = threadIdx.x & 31;
  const int m0 = (blockIdx.x * 8 + (threadIdx.x >> 5)) * 16;
  if (m0 >= M) return;

  const int r16 = lane & 15;
  const int khalf = (lane >> 4) << 1;  // 0 or 2

  v8f acc[NTILES];
#pragma unroll
  for (int t = 0; t < NTILES; ++t)
#pragma unroll
    for (int i = 0; i < 8; ++i) acc[t][i] = 0.0f;

  const float* aBase = A + (size_t)(m0 + r16) * K + khalf;  // 8B-aligned
  for (int k0 = 0; k0 < K; k0 += 4) {
    v2f a = *(const v2f*)(aBase + k0);                     // global_load_b64
    const float* bRow = ldsB + ((k0 + khalf) >> 1) * (2 * NP);
#pragma unroll
    for (int t = 0; t < NTILES; ++t) {
      v2f b = *(const v2f*)(bRow + (t * 16 + r16) * 2);    // ds_load_b64
      acc[t] = __builtin_amdgcn_wmma_f32_16x16x4_f32(
          false, a, false, b, (short)0, acc[t], false, false);
    }
  }

  const int mbase = m0 + ((lane >> 4) << 3);
#pragma unroll
  for (int t = 0; t < NTILES; ++t) {
    const int col = t * 16 + r16;
    if (col < N) {
      const float bv = bias ? bias[col] : 0.0f;
#pragma unroll
      for (int i = 0; i < 8; ++i) {
        C[(size_t)(mbase + i) * N + col] = acc[t][i] + bv;
      }
    }
  }
}

// ---------------------------------------------------------------------------
// Embedding layer 1: out[r,c] = points[r,:3] @ w[3,64] + b[c]
// ---------------------------------------------------------------------------
__global__ void embed1_kernel(const float* __restrict__ pts,
                              const float* __restrict__ w,
                              const float* __restrict__ bias,
                              float* __restrict__ out, int rows) {
  int idx = blockIdx.x * blockDim.x + threadIdx.x;
  if (idx >= rows * 64) return;
  int r = idx >> 6;
  int c = idx & 63;
  float p0 = pts[r * 3 + 0], p1 = pts[r * 3 + 1], p2 = pts[r * 3 + 2];
  out[idx] = fmaf(p0, w[c], fmaf(p1, w[64 + c], fmaf(p2, w[128 + c], bias[c])));
}

// ---------------------------------------------------------------------------
// BatchNorm (training mode) stats: per-channel sum & sum-of-squares.
// stats[0..C-1] = sum, stats[C..2C-1] = sumsq.  Pre-zeroed with memset.
// blockDim = 256; C divides 256 (C is 64 or 128 here).
// ---------------------------------------------------------------------------
__global__ void bn_stats_kernel(const float* __restrict__ x,
                                float* __restrict__ stats, int rows, int C) {
  const int rpi = 256 / C;
  const int c = threadIdx.x % C;
  const int rsub = threadIdx.x / C;
  const int rowsPerBlock = (rows + gridDim.x - 1) / gridDim.x;
  const int r0 = blockIdx.x * rowsPerBlock;
  int r1 = r0 + rowsPerBlock;
  if (r1 > rows) r1 = rows;

  float s = 0.0f, s2 = 0.0f;
  for (int r = r0 + rsub; r < r1; r += rpi) {
    float v = x[(size_t)r * C + c];
    s += v;
    s2 += v * v;
  }
  __shared__ float sh[512];
  sh[threadIdx.x] = s;
  sh[256 + threadIdx.x] = s2;
  __syncthreads();
  if (threadIdx.x < C) {
    float ts = 0.0f, ts2 = 0.0f;
    for (int t = threadIdx.x; t < 256; t += C) {
      ts += sh[t];
      ts2 += sh[256 + t];
    }
    atomicAdd(&stats[c], ts);
    atomicAdd(&stats[C + c], ts2);
  }
}

// y = relu((x - mean) * rsqrt(var + eps) * g + b)
__global__ void bn_apply_relu_kernel(const float* __restrict__ x,
                                     const float* __restrict__ stats,
                                     const float* __restrict__ g,
                                     const float* __restrict__ bta,
                                     float* __restrict__ y, int rows, int C) {
  int idx = blockIdx.x * blockDim.x + threadIdx.x;
  if (idx >= rows * C) return;
  int c = idx % C;
  float cnt = (float)rows;
  float m = stats[c] / cnt;
  float var = stats[C + c] / cnt - m * m;
  float r = rsqrtf(var + 1e-5f);
  float v = (x[idx] - m) * r * g[c] + bta[c];
  y[idx] = fmaxf(v, 0.0f);
}

// ---------------------------------------------------------------------------
// Brute-force kNN (k=16, self excluded). One query point per thread,
// candidates staged in LDS as float4, register-resident top-16 with
// fully-unrolled constant-index arrays (stays in VGPRs).
// knn[p*17 + 0..15] = 16 nearest, knn[p*17 + 16] = self.
// ---------------------------------------------------------------------------
#define KNN_CHUNK 1024

__global__ void knn_kernel(const float* __restrict__ pos,
                           int* __restrict__ knn, int Nn) {
  __shared__ float4 sh[KNN_CHUNK];
  const int gid = blockIdx.x * blockDim.x + threadIdx.x;
  const int b = gid / Nn;
  const int i = gid % Nn;
  const float* pb = pos + (size_t)b * Nn * 3;
  const float px = pb[i * 3 + 0];
  const float py = pb[i * 3 + 1];
  const float pz = pb[i * 3 + 2];

  float bd[16];
  int bi[16];
#pragma unroll
  for (int t = 0; t < 16; ++t) {
    bd[t] = 3.0e38f;
    bi[t] = 0;
  }
  float worst = 3.0e38f;
  int wslot = 0;

  for (int c0 = 0; c0 < Nn; c0 += KNN_CHUNK) {
    for (int t = threadIdx.x; t < KNN_CHUNK; t += blockDim.x) {
      int j = c0 + t;
      sh[t] = make_float4(pb[j * 3 + 0], pb[j * 3 + 1], pb[j * 3 + 2], 0.0f);
    }
    __syncthreads();
    for (int t = 0; t < KNN_CHUNK; ++t) {
      float4 pj = sh[t];  // uniform index -> LDS broadcast
      int j = c0 + t;
      float dx = px - pj.x;
      float dy = py - pj.y;
      float dz = pz - pj.z;
      float d2 = fmaf(dx, dx, fmaf(dy, dy, dz * dz));
      if (d2 < worst && j != i) {
#pragma unroll
        for (int s = 0; s < 16; ++s)
          if (s == wslot) {
            bd[s] = d2;
            bi[s] = j;
          }
        worst = bd[0];
        wslot = 0;
#pragma unroll
        for (int s = 1; s < 16; ++s)
          if (bd[s] > worst) {
            worst = bd[s];
            wslot = s;
          }
      }
    }
    __syncthreads();
  }

  int* out = knn + (size_t)gid * 17;
#pragma unroll
  for (int t = 0; t < 16; ++t) out[t] = bi[t];
  out[16] = i;
}

// ---------------------------------------------------------------------------
// PointTransformerConv attention + aggregation.
// One block (128 threads) per point; thread c owns channel c, so the 17-way
// softmax is sequential per-thread in registers (no cross-lane ops).
// Neighbor indices + positions staged in LDS by the first 17 threads.
//   delta_n = (pos_i - pos_j) @ pos_w + pos_b
//   attn    = softmax_n(a_dst_i - a_src_j + delta_n)
//   out_i   = sum_n attn_n * (v_j + delta_n)
// ---------------------------------------------------------------------------
#define NBR 17

__global__ void pt_conv_kernel(const float* __restrict__ v,
                               const float* __restrict__ asrc,
                               const float* __restrict__ adst,
                               const float* __restrict__ pos,
                               const int* __restrict__ knn,
                               const float* __restrict__ pos_w,
                               const float* __restrict__ pos_b,
                               float* __restrict__ out, int Nn, int C) {
  const int p = blockIdx.x;  // global point id
  const int b = p / Nn;
  const int i = p % Nn;
  const int c = threadIdx.x;
  const float* pb = pos + (size_t)b * Nn * 3;

  __shared__ int sj[NBR];
  __shared__ float4 spos[NBR];
  if (threadIdx.x < NBR) {
    int j = knn[(size_t)p * NBR + threadIdx.x];
    sj[threadIdx.x] = j;
    spos[threadIdx.x] =
        make_float4(pb[j * 3 + 0], pb[j * 3 + 1], pb[j * 3 + 2], 0.0f);
  }
  __syncthreads();

  const float pix = pb[i * 3 + 0];
  const float piy = pb[i * 3 + 1];
  const float piz = pb[i * 3 + 2];
  const float w0 = pos_w[0 * C + c];
  const float w1 = pos_w[1 * C + c];
  const float w2 = pos_w[2 * C + c];
  const float wb = pos_b[c];
  const float ad = adst[(size_t)p * C + c];

  float e[NBR], dl[NBR];
  float emax = -3.0e38f;
#pragma unroll
  for (int n = 0; n < NBR; ++n) {
    float4 pj = spos[n];
    size_t jg = (size_t)b * Nn + sj[n];
    float dx = pix - pj.x;
    float dy = piy - pj.y;
    float dz = piz - pj.z;
    float delta = fmaf(dx, w0, fmaf(dy, w1, fmaf(dz, w2, wb)));
    dl[n] = delta;
    float ev = ad - asrc[jg * C + c] + delta;
    e[n] = ev;
    emax = fmaxf(emax, ev);
  }
  float esum = 0.0f;
#pragma unroll
  for (int n = 0; n < NBR; ++n) {
    e[n] = __expf(e[n] - emax);
    esum += e[n];
  }
  const float inv = 1.0f / esum;
  float acc = 0.0f;
#pragma unroll
  for (int n = 0; n < NBR; ++n) {
    size_t jg = (size_t)b * Nn + sj[n];
    acc = fmaf(e[n] * inv, v[jg * C + c] + dl[n], acc);
  }
  out[(size_t)p * C + c] = acc;
}

// ---------------------------------------------------------------------------
// Host side
// ---------------------------------------------------------------------------
template <int NTILES>
static void launch_gemm(const float* A, const float* B, const float* bias,
                        float* C, int M, int N, int K, hipStream_t s) {
  int blocks = (M + 127) / 128;  // 8 waves x 16 rows per block
  size_t shmem = (size_t)K * NTILES * 16 * sizeof(float);
  wmma_gemm_f32<NTILES><<<blocks, 256, shmem, s>>>(A, B, bias, C, M, N, K);
}

extern "C" void kernel_launch(void* const* d_in, const int* in_sizes, int n_in,
                              void* d_out, int out_size, void* d_ws,
                              size_t ws_size, hipStream_t stream) {
  const int Bc = 4, Nn = 8192, R = Bc * Nn;

  const float* points   = (const float*)d_in[0];
  const float* embed_w1 = (const float*)d_in[1];
  const float* embed_b1 = (const float*)d_in[2];
  const float* bn1_g    = (const float*)d_in[3];
  const float* bn1_b    = (const float*)d_in[4];
  const float* embed_w2 = (const float*)d_in[5];
  const float* embed_b2 = (const float*)d_in[6];
  const float* bne_g    = (const float*)d_in[7];
  const float* bne_b    = (const float*)d_in[8];
  const float* l0_lin   = (const float*)d_in[9];
  const float* l0_src   = (const float*)d_in[10];
  const float* l0_dst   = (const float*)d_in[11];
  const float* l0_pos_w = (const float*)d_in[12];
  const float* l0_pos_b = (const float*)d_in[13];
  const float* l0_bn_g  = (const float*)d_in[14];
  const float* l0_bn_b  = (const float*)d_in[15];
  const float* l1_lin   = (const float*)d_in[16];
  const float* l1_src   = (const float*)d_in[17];
  const float* l1_dst   = (const float*)d_in[18];
  const float* l1_pos_w = (const float*)d_in[19];
  const float* l1_pos_b = (const float*)d_in[20];
  const float* l1_bn_g  = (const float*)d_in[21];
  const float* l1_bn_b  = (const float*)d_in[22];
  const float* dec_w1   = (const float*)d_in[23];
  const float* dec_b1   = (const float*)d_in[24];
  const float* dec_bn_g = (const float*)d_in[25];
  const float* dec_bn_b = (const float*)d_in[26];
  const float* dec_w2   = (const float*)d_in[27];
  const float* dec_b2   = (const float*)d_in[28];

  float* ws = (float*)d_ws;
  size_t off = 0;
  float* f_x0  = ws + off; off += (size_t)R * 64;   // embedding output (64ch)
  float* f_y   = ws + off; off += (size_t)R * 128;  // generic pre-BN buffer
  float* f_v   = ws + off; off += (size_t)R * 128;  // value / temp
  float* f_src = ws + off; off += (size_t)R * 128;
  float* f_dst = ws + off; off += (size_t)R * 128;
  float* f_x1  = ws + off; off += (size_t)R * 128;  // layer-1 input
  float* stats = ws + off; off += 256;
  int* knn = (int*)(ws + off);

  const int e64 = (R * 64 + 255) / 256;
  const int e128 = (R * 128 + 255) / 256;

  // ---- Embedding MLP: Linear(3->64) -> BN -> ReLU -> Linear(64->64) -> BN -> ReLU
  embed1_kernel<<<e64, 256, 0, stream>>>(points, embed_w1, embed_b1, f_y, R);
  hipMemsetAsync(stats, 0, 256 * sizeof(float), stream);
  bn_stats_kernel<<<64, 256, 0, stream>>>(f_y, stats, R, 64);
  bn_apply_relu_kernel<<<e64, 256, 0, stream>>>(f_y, stats, bn1_g, bn1_b, f_v, R, 64);

  launch_gemm<4>(f_v, embed_w2, embed_b2, f_y, R, 64, 64, stream);
  hipMemsetAsync(stats, 0, 256 * sizeof(float), stream);
  bn_stats_kernel<<<64, 256, 0, stream>>>(f_y, stats, R, 64);
  bn_apply_relu_kernel<<<e64, 256, 0, stream>>>(f_y, stats, bne_g, bne_b, f_x0, R, 64);

  // ---- kNN graph (k=16 + self loop)
  knn_kernel<<<R / 256, 256, 0, stream>>>(points, knn, Nn);

  // ---- PointTransformerConv layer 0 (64 -> 128)
  launch_gemm<8>(f_x0, l0_lin, nullptr, f_v, R, 128, 64, stream);
  launch_gemm<8>(f_x0, l0_src, nullptr, f_src, R, 128, 64, stream);
  launch_gemm<8>(f_x0, l0_dst, nullptr, f_dst, R, 128, 64, stream);
  pt_conv_kernel<<<R, 128, 0, stream>>>(f_v, f_src, f_dst, points, knn,
                                        l0_pos_w, l0_pos_b, f_y, Nn, 128);
  hipMemsetAsync(stats, 0, 256 * sizeof(float), stream);
  bn_stats_kernel<<<64, 256, 0, stream>>>(f_y, stats, R, 128);
  bn_apply_relu_kernel<<<e128, 256, 0, stream>>>(f_y, stats, l0_bn_g, l0_bn_b,
                                                 f_x1, R, 128);

  // ---- PointTransformerConv layer 1 (128 -> 128)
  launch_gemm<8>(f_x1, l1_lin, nullptr, f_v, R, 128, 128, stream);
  launch_gemm<8>(f_x1, l1_src, nullptr, f_src, R, 128, 128, stream);
  launch_gemm<8>(f_x1, l1_dst, nullptr, f_dst, R, 128, 128, stream);
  pt_conv_kernel<<<R, 128, 0, stream>>>(f_v, f_src, f_dst, points, knn,
                                        l1_pos_w, l1_pos_b, f_y, Nn, 128);
  hipMemsetAsync(stats, 0, 256 * sizeof(float), stream);
  bn_stats_kernel<<<64, 256, 0, stream>>>(f_y, stats, R, 128);
  bn_apply_relu_kernel<<<e128, 256, 0, stream>>>(f_y, stats, l1_bn_g, l1_bn_b,
                                                 f_v, R, 128);

  // ---- Decoder: Linear(128->128)+BN+ReLU, Linear(128->13)
  launch_gemm<8>(f_v, dec_w1, dec_b1, f_src, R, 128, 128, stream);
  hipMemsetAsync(stats, 0, 256 * sizeof(float), stream);
  bn_stats_kernel<<<64, 256, 0, stream>>>(f_src, stats, R, 128);
  bn_apply_relu_kernel<<<e128, 256, 0, stream>>>(f_src, stats, dec_bn_g,
                                                 dec_bn_b, f_dst, R, 128);
  launch_gemm<1>(f_dst, dec_w2, dec_b2, (float*)d_out, R, 13, 128, stream);
}